// VectorQuantizer_65893388255954
// MI455X (gfx1250) — compile-verified
//
#include <hip/hip_runtime.h>
#include <hip/hip_bf16.h>

typedef __attribute__((ext_vector_type(2))) float v2f;
typedef __attribute__((ext_vector_type(4))) float v4f;
typedef __attribute__((ext_vector_type(8))) float v8f;
typedef __attribute__((ext_vector_type(4))) unsigned int u32x4;
typedef __attribute__((ext_vector_type(8))) int i32x8;
typedef __attribute__((ext_vector_type(4))) int i32x4;

#define ROWS        65536          // 16*4096
#define DIMS        64
#define NEMB        1024
#define TOPK        8
#define NC          64             // codebook chunk width (columns)
#define NCHUNK      (NEMB / NC)    // 16
#define BLK_WAVES   4
#define BLK_THREADS 128
#define ROWS_PER_BLK 64
#define NBLK        (ROWS / ROWS_PER_BLK)   // 1024 blocks
#define EMB_STRIDE  72             // padded LDS row stride for emb chunk (TDM pad 64+8)
#define DIST_STRIDE 66             // padded LDS row stride for dist tile
#define OUT_Q_ELEMS (ROWS * TOPK * DIMS)    // 33554432

// workspace layout (floats)
#define WS_EMBT  0                       // 1024*64 transposed codebook
#define WS_ENORM (NEMB * DIMS)           // 1024 column norms
#define WS_PART  (WS_ENORM + NEMB)       // 1024 per-block loss partials

__device__ __forceinline__ void topk_insert(float d, int idx, float s[TOPK], int si[TOPK]) {
    if (d < s[TOPK - 1]) {
        s[TOPK - 1] = d; si[TOPK - 1] = idx;
#pragma unroll
        for (int t = TOPK - 1; t > 0; --t) {
            if (s[t] < s[t - 1]) {
                float tf = s[t]; s[t] = s[t - 1]; s[t - 1] = tf;
                int   ti = si[t]; si[t] = si[t - 1]; si[t - 1] = ti;
            }
        }
    }
}

// Kernel 1: transpose codebook (64x1024 -> 1024x64) and compute per-column norms.
__global__ void vq_prep(const float* __restrict__ emb, float* __restrict__ ws) {
    int n = blockIdx.x * blockDim.x + threadIdx.x;
    if (n >= NEMB) return;
    float s = 0.f;
    float* embT = ws + WS_EMBT;
#pragma unroll 8
    for (int k = 0; k < DIMS; ++k) {
        float v = emb[k * NEMB + n];
        embT[n * DIMS + k] = v;
        s += v * v;
    }
    ws[WS_ENORM + n] = s;
}

// Kernel 2: distances via V_WMMA_F32_16X16X4_F32 with TDM-staged codebook chunks,
// streaming top-8, gather, loss partials.
__global__ void __launch_bounds__(BLK_THREADS)
vq_main(const float* __restrict__ x, const float* __restrict__ emb,
        float* __restrict__ ws, float* __restrict__ out) {
    __shared__ float lds_emb[DIMS * EMB_STRIDE];                 // 18432 B
    __shared__ float lds_dist[BLK_WAVES * 16 * DIST_STRIDE];     // 16896 B
    __shared__ int   lds_idx[BLK_WAVES * 16 * TOPK];             //  2048 B
    __shared__ float lds_wsum[BLK_WAVES];

    const int tid  = threadIdx.x;
    const int w    = tid >> 5;
    const int lane = tid & 31;
    const int m    = lane & 15;     // row within wave's 16-row tile
    const int hf   = lane >> 4;     // half of wave
    const int row_base = blockIdx.x * ROWS_PER_BLK + w * 16;

    const float* embT  = ws + WS_EMBT;
    const float* enorm = ws + WS_ENORM;

    // ---- load A fragments (16 rows x 64 cols), pre-scaled by -2 ----
    // 16x16x4 f32 A layout: lanes 0-15 M=lane, VGPR0=K0/VGPR1=K1; lanes 16-31 K2/K3.
    v2f afr[16];
    {
        const float* xr = x + (size_t)(row_base + m) * DIMS;
#pragma unroll
        for (int kk = 0; kk < 16; ++kk) {
            v2f a = *(const v2f*)(xr + kk * 4 + 2 * hf);
            afr[kk] = a * -2.0f;
        }
    }

    float s[TOPK]; int si[TOPK];
#pragma unroll
    for (int i = 0; i < TOPK; ++i) { s[i] = 3.402823466e+38f; si[i] = 0; }

    for (int c = 0; c < NCHUNK; ++c) {
        const int n0 = c * NC;

        // ---- stage codebook chunk (64 rows x 64 cols) into LDS ----
#if defined(__gfx1250__) && __has_builtin(__builtin_amdgcn_tensor_load_to_lds) && \
    __has_builtin(__builtin_amdgcn_s_wait_tensorcnt)
        // Tensor Data Mover: one descriptor moves the whole 64x64 f32 tile.
        // D# group0: count=1, lds_addr, global_addr, type=2 (ISA 08 §8.3)
        // D# group1: data_size=4B, pad_enable, pad_interval=5 (64 DW), pad_amount=7 (8 DW)
        //            => LDS row stride 72 floats; tensor_dim0=1024, stride=1024,
        //            tensor_dim1=64, tile_dim0=64, tile_dim1=64 (ISA 08 §8.4)
        if (tid < 32) {
            unsigned long long ga = (unsigned long long)(const void*)(emb + n0);
            unsigned int       la = (unsigned int)(unsigned long long)(const void*)lds_emb;
            u32x4 g0 = { 1u, la, (unsigned int)ga,
                         (unsigned int)(ga >> 32) | (2u << 30) };
            i32x8 g1 = { (int)((2u << 16) | (1u << 20) | (5u << 22) | (7u << 25)),
                         (int)(1024u << 16),   // tensor_dim0 = 1024 (bits 79:48)
                         (64 << 16),           // tensor_dim1 = 64  (bits 111:80)
                         (64 << 16),           // tile_dim0  = 64  (bits 127:112)
                         64,                   // tile_dim1  = 64  (bits 143:128)
                         1024,                 // tensor_dim0_stride (bits 207:160)
                         0, 0 };
            i32x4 gz4 = { 0, 0, 0, 0 };
            i32x8 gz8 = { 0, 0, 0, 0, 0, 0, 0, 0 };
            // 6-arg form (clang-23 / therock-10.0 headers): groups 0..3 + extra + cpol
            __builtin_amdgcn_tensor_load_to_lds(g0, g1, gz4, gz4, gz8, 0);
            __builtin_amdgcn_s_wait_tensorcnt(0);   // drain before releasing barrier
        }
        if (c + 1 < NCHUNK)
            __builtin_prefetch(emb + (size_t)(tid >> 1) * NEMB + n0 + NC + (tid & 1) * 32, 0, 1);
#else
        {
            const int k  = tid >> 1;
            const int ph = tid & 1;
            const float* g = emb + (size_t)k * NEMB + n0 + ph * 32;
            float* l = lds_emb + k * EMB_STRIDE + ph * 32;
#pragma unroll
            for (int i = 0; i < 8; ++i)
                *(v4f*)(l + i * 4) = *(const v4f*)(g + i * 4);
            if (c + 1 < NCHUNK) __builtin_prefetch(g + NC, 0, 1);
        }
#endif
        __syncthreads();

        // ---- D = (-2x)*E + ||e||^2 over 4 N-tiles of 16 ----
#pragma unroll
        for (int nt = 0; nt < NC / 16; ++nt) {
            const int coll = nt * 16 + m;
            const float en = enorm[n0 + coll];
            v8f acc = {en, en, en, en, en, en, en, en};
#pragma unroll
            for (int kk = 0; kk < 16; ++kk) {
                const int kb = kk * 4 + 2 * hf;   // B: lanes 0-15 rows K0/K1, lanes 16-31 K2/K3
                v2f b;
                b.x = lds_emb[(kb + 0) * EMB_STRIDE + coll];
                b.y = lds_emb[(kb + 1) * EMB_STRIDE + coll];
                acc = __builtin_amdgcn_wmma_f32_16x16x4_f32(
                          false, afr[kk], false, b, (short)0, acc, false, false);
            }
#pragma unroll
            for (int r = 0; r < 8; ++r)
                lds_dist[(w * 16 + r + 8 * hf) * DIST_STRIDE + coll] = acc[r];
        }
        __syncthreads();

        // ---- streaming top-8: lane pair (m, m+16) scans its 32 candidates ----
        {
            const float* dr = lds_dist + (w * 16 + m) * DIST_STRIDE + hf * 32;
            const int nb = n0 + hf * 32;
            for (int j = 0; j < 32; ++j)
                topk_insert(dr[j], nb + j, s, si);
        }
        __syncthreads();
    }

    // ---- merge the two half-lists of each row ----
    {
        float pd[TOPK]; int pi[TOPK];
#pragma unroll
        for (int t = 0; t < TOPK; ++t) {
            pd[t] = __shfl_xor(s[t], 16, 32);
            pi[t] = __shfl_xor(si[t], 16, 32);
        }
#pragma unroll
        for (int t = 0; t < TOPK; ++t) topk_insert(pd[t], pi[t], s, si);
    }
    if (hf == 0) {
#pragma unroll
        for (int k = 0; k < TOPK; ++k) lds_idx[(w * 16 + m) * TOPK + k] = si[k];
    }
    __syncthreads();

    // ---- gather quantized vectors (coalesced via embT), write out, loss partial ----
    float accl = 0.f;
    for (int p = 0; p < 16 * TOPK; ++p) {
        const int rr  = p >> 3, k = p & 7;
        const int idx = lds_idx[(w * 16 + rr) * TOPK + k];
        const int row = row_base + rr;
        const float* qv = embT + (size_t)idx * DIMS;
        const float* xv = x + (size_t)row * DIMS;
        float q0 = qv[lane], q1 = qv[lane + 32];
        float x0 = xv[lane], x1 = xv[lane + 32];
        float* op = out + ((size_t)row * TOPK + k) * DIMS;
        op[lane] = q0; op[lane + 32] = q1;
        float d0 = q0 - x0, d1 = q1 - x1;
        accl += d0 * d0 + d1 * d1;
    }
#pragma unroll
    for (int o = 16; o > 0; o >>= 1) accl += __shfl_xor(accl, o, 32);
    if (lane == 0) lds_wsum[w] = accl;
    __syncthreads();
    if (tid == 0)
        ws[WS_PART + blockIdx.x] = lds_wsum[0] + lds_wsum[1] + lds_wsum[2] + lds_wsum[3];
}

// Kernel 3: deterministic final loss reduction.
__global__ void vq_reduce(const float* __restrict__ ws, float* __restrict__ out) {
    __shared__ float sm[256];
    const int t = threadIdx.x;
    float s = 0.f;
    for (int i = t; i < NBLK; i += 256) s += ws[WS_PART + i];
    sm[t] = s; __syncthreads();
    for (int o = 128; o > 0; o >>= 1) {
        if (t < o) sm[t] += sm[t + o];
        __syncthreads();
    }
    if (t == 0) {
        float mean = sm[0] * (1.0f / (float)OUT_Q_ELEMS);
        out[OUT_Q_ELEMS] = 0.25f * mean + mean;   // BETA*commitment + codebook
    }
}

extern "C" void kernel_launch(void* const* d_in, const int* in_sizes, int n_in,
                              void* d_out, int out_size, void* d_ws, size_t ws_size,
                              hipStream_t stream) {
    const float* x   = (const float*)d_in[0];
    const float* emb = (const float*)d_in[1];
    float* out = (float*)d_out;
    float* ws  = (float*)d_ws;

    vq_prep<<<dim3(NEMB / 256), dim3(256), 0, stream>>>(emb, ws);
    vq_main<<<dim3(NBLK), dim3(BLK_THREADS), 0, stream>>>(x, emb, ws, out);
    vq_reduce<<<dim3(1), dim3(256), 0, stream>>>(ws, out);
}